// EDSCWPool2d_309237645645
// MI455X (gfx1250) — compile-verified
//
#include <hip/hip_runtime.h>
#include <hip/hip_bf16.h>
#include <float.h>

// Problem constants (from reference): B=16 C=96 H=W=224, kernel 2x2 stride 2
// -> oH=oW=112. Output pairs = 16*96*112*56 = 9,633,792 = 1792*256*21 exactly.
//
// Layout algebra (NCHW, contiguous):
//   pair index p -> output floats {2p, 2p+1}           (output is flat)
//   input row0 offset(p) = 8p - 4*(p % 56)  floats     (row1 = +224 floats)
// Grid stride 1792*256 = 56*8192 is divisible by 56, so (p % 56) is
// loop-invariant per thread and both addresses advance by constant strides.
#define TPB    256
#define BLOCKS 1792
#define NITER  21

typedef __attribute__((ext_vector_type(4))) float v4f;
typedef __attribute__((ext_vector_type(2))) float v2f;

// Per-window EDSCW: m = mean(v0..v3); w_i = exp(2 v_i m / (v_i^2 + m^2));
// out = sum(w_i v_i) / sum(w_i); nan_to_num at the end.
__device__ __forceinline__ float edscw4(float v0, float v1, float v2, float v3) {
    const float m  = 0.25f * ((v0 + v1) + (v2 + v3));
    const float m2 = m * m;
    float vv[4] = {v0, v1, v2, v3};
    float ws = 0.0f, wv = 0.0f;
#pragma unroll
    for (int k = 0; k < 4; ++k) {
        const float v   = vv[k];
        const float dsc = (2.0f * v * m) * __builtin_amdgcn_rcpf(v * v + m2);
        const float w   = __expf(dsc);          // v_exp_f32
        ws += w;
        wv = __builtin_fmaf(w, v, wv);
    }
    float r = wv * __builtin_amdgcn_rcpf(ws);
    // nan_to_num: NaN -> 0, +/-inf -> +/-FLT_MAX
    r = (r != r) ? 0.0f : r;
    r = fminf(fmaxf(r, -FLT_MAX), FLT_MAX);
    return r;
}

// Issue the two async b128 loads (row0 + row1 of this thread's 2 windows)
// into LDS at byte address ldsA / ldsA+16. Row pitch = 224 floats = 896 B,
// folded into the instruction offset so both share one 64-bit address VGPR.
__device__ __forceinline__ void prefetch_pair(const float* g, uint32_t ldsA) {
    const unsigned long long ga = (unsigned long long)(uintptr_t)g;
    asm volatile("global_load_async_to_lds_b128 %0, %1, off"
                 :: "v"(ldsA), "v"(ga) : "memory");
    asm volatile("global_load_async_to_lds_b128 %0, %1, off offset:896"
                 :: "v"(ldsA + 16u), "v"(ga) : "memory");
}

__global__ __launch_bounds__(TPB)
void edscw_pool_kernel(const float* __restrict__ x, float* __restrict__ out) {
    __shared__ v4f sb[2][TPB * 2];          // 2 x 8KB double buffer

    const uint32_t tid    = threadIdx.x;
    const uint32_t base   = blockIdx.x * TPB + tid;
    const uint32_t stride = BLOCKS * TPB;   // pairs per grid iteration
    const uint32_t ow2    = base % 56u;     // loop-invariant (stride % 56 == 0)

    // Generic LDS pointers are {aperture_hi32, lds_offset32}: low 32 bits are
    // the byte offset the async-to-LDS instruction wants.
    uint32_t lds[2];
    lds[0] = (uint32_t)(uintptr_t)(&sb[0][tid * 2u]);
    lds[1] = (uint32_t)(uintptr_t)(&sb[1][tid * 2u]);

    const float* g = x + ((size_t)base * 8u - (size_t)ow2 * 4u); // tile 0 row0
    v2f*         o = (v2f*)out + base;                           // pair p slot
    const size_t gstep = (size_t)stride * 8u;                    // floats/iter
    const size_t ostep = (size_t)stride;                         // pairs/iter

    // Prime the pipeline with tile 0.
    prefetch_pair(g, lds[0]);

#pragma unroll 1
    for (uint32_t i = 0; i < NITER; ++i) {
        if (i + 1u < NITER) {
            // Prefetch next tile into the other buffer, then wait until only
            // its 2 async ops remain outstanding (async loads complete in
            // order, so <=2 outstanding means the current tile has landed).
            prefetch_pair(g + gstep, lds[(i + 1u) & 1u]);
            asm volatile("s_wait_asynccnt 0x2" ::: "memory");
        } else {
            asm volatile("s_wait_asynccnt 0x0" ::: "memory");
        }

        const v4f r0 = sb[i & 1u][tid * 2u + 0u];   // row0: cols 4*ow2..+3
        const v4f r1 = sb[i & 1u][tid * 2u + 1u];   // row1

        v2f res;
        res.x = edscw4(r0.x, r0.y, r1.x, r1.y);
        res.y = edscw4(r0.z, r0.w, r1.z, r1.w);
        __builtin_nontemporal_store(res, o);

        g += gstep;
        o += ostep;
    }
}

extern "C" void kernel_launch(void* const* d_in, const int* in_sizes, int n_in,
                              void* d_out, int out_size, void* d_ws, size_t ws_size,
                              hipStream_t stream) {
    (void)in_sizes; (void)n_in; (void)out_size; (void)d_ws; (void)ws_size;
    const float* x = (const float*)d_in[0];
    float* out     = (float*)d_out;
    edscw_pool_kernel<<<BLOCKS, TPB, 0, stream>>>(x, out);
}